// ChebNet_489626272291
// MI455X (gfx1250) — compile-verified
//
#include <hip/hip_runtime.h>

// ---------------------------------------------------------------------------
// ChebNet-GL-NL fused pipeline for MI455X (gfx1250, wave32, WMMA f32 16x16x4)
// ---------------------------------------------------------------------------

#define NN   307            // graph nodes
#define NP   320            // padded nodes (20 tiles of 16)
#define HH   16             // hidden width
#define CIN  5              // input channels
#define KCH  6              // Chebyshev orders
#define BB   512            // batch
#define ROWS (BB*NN)        // 157184 flattened (b,n) rows  (== 16*9824)
#define ST   17             // LDS row stride (bank-conflict padding)

typedef __attribute__((ext_vector_type(2))) float v2f;
typedef __attribute__((ext_vector_type(8))) float v8f;

__device__ __forceinline__ v8f zero8() {
  v8f z = {0.f,0.f,0.f,0.f,0.f,0.f,0.f,0.f};
  return z;
}

// D = A(16x4) * B(4x16) + C   (fp32 WMMA)
__device__ __forceinline__ v8f wmma4(v2f a, v2f b, v8f c) {
  return __builtin_amdgcn_wmma_f32_16x16x4_f32(false, a, false, b, (short)0, c,
                                               false, false);
}

// A fragment of a 16x4 f32 tile at p (row stride ld floats)
// lane l: m = l&15 ; holds A[m][ks], A[m][ks+1] with ks = 2*(l>=16)
__device__ __forceinline__ v2f fragA(const float* p, int ld) {
  const int l  = threadIdx.x & 31;
  const int m  = l & 15;
  const int ks = (l >> 4) << 1;
  v2f a;
  a.x = p[m * ld + ks];
  a.y = p[m * ld + ks + 1];
  return a;
}

// B fragment of a 4x16 f32 tile at p (row stride ld floats)
// lane l: n = l&15 ; holds B[ks][n], B[ks+1][n]
__device__ __forceinline__ v2f fragB(const float* p, int ld) {
  const int l  = threadIdx.x & 31;
  const int n  = l & 15;
  const int ks = (l >> 4) << 1;
  v2f b;
  b.x = p[ks * ld + n];
  b.y = p[(ks + 1) * ld + n];
  return b;
}

// C/D fragment store / load, tile at p (row stride ld)
// lane l, vgpr v: row = v + 8*(l>=16), col = l&15
__device__ __forceinline__ void stC(float* p, int ld, v8f c) {
  const int l  = threadIdx.x & 31;
  const int n  = l & 15;
  const int mb = (l >> 4) << 3;
#pragma unroll
  for (int v = 0; v < 8; ++v) p[(mb + v) * ld + n] = c[v];
}

__device__ __forceinline__ v8f ldC(const float* p, int ld) {
  const int l  = threadIdx.x & 31;
  const int n  = l & 15;
  const int mb = (l >> 4) << 3;
  v8f c;
#pragma unroll
  for (int v = 0; v < 8; ++v) c[v] = p[(mb + v) * ld + n];
  return c;
}

__device__ __forceinline__ v8f tanh8(v8f x) {
  v8f r;
#pragma unroll
  for (int v = 0; v < 8; ++v) r[v] = tanhf(x[v]);
  return r;
}

__device__ __forceinline__ float sigm(float x) { return 1.f / (1.f + __expf(-x)); }

// ---------------------------------------------------------------------------
// Stage 0: graph / Laplacian / Chebyshev polynomials (zero-padded to 320x320)
// ---------------------------------------------------------------------------

__global__ void graph_kernel(const float* __restrict__ emb, float* __restrict__ G) {
  int i = blockIdx.x * blockDim.x + threadIdx.x;
  if (i >= NP * NP) return;
  int n = i / NP, m = i - n * NP;
  float g = 0.f;
  if (n < NN && m < NN) {
    float dn = 0.f, dm = 0.f, dd = 0.f;
#pragma unroll
    for (int c = 0; c < 10; ++c) {
      float a = emb[n * 10 + c], b = emb[m * 10 + c];
      dd += a * b; dn += a * a; dm += b * b;
    }
    g = dd * rsqrtf(dn) * rsqrtf(dm);
  }
  G[i] = g;
}

__global__ void degree_kernel(const float* __restrict__ G, float* __restrict__ dinv) {
  int n = threadIdx.x;
  if (n >= NP) return;
  float r = 0.f;
  if (n < NN) {
    float s = 0.f;
    for (int m = 0; m < NN; ++m) s += G[n * NP + m];
    r = rsqrtf(s);
  }
  dinv[n] = r;
}

__global__ void lap_kernel(const float* __restrict__ G, const float* __restrict__ dinv,
                           float* __restrict__ Lp, float* __restrict__ T0,
                           float* __restrict__ T1) {
  int i = blockIdx.x * blockDim.x + threadIdx.x;
  if (i >= NP * NP) return;
  int n = i / NP, m = i - n * NP;
  float l = 0.f, t0 = 0.f;
  if (n < NN && m < NN) {
    l  = ((n == m) ? 1.f : 0.f) - dinv[n] * G[i] * dinv[m];
    t0 = (n == m) ? 1.f : 0.f;
  }
  Lp[i] = l; T0[i] = t0; T1[i] = l;
}

// Tk = 2 * Lp @ Tm1 - Tm2   (padded zeros make the 320-length dot exact)
__global__ void cheb_kernel(const float* __restrict__ Lp, const float* __restrict__ Tm1,
                            const float* __restrict__ Tm2, float* __restrict__ Tk) {
  int i = blockIdx.x * blockDim.x + threadIdx.x;
  if (i >= NP * NP) return;
  int n = i / NP, m = i - n * NP;
  float v = 0.f;
  if (n < NN && m < NN) {
    float s = 0.f;
    for (int j = 0; j < NP; ++j) s += Lp[n * NP + j] * Tm1[j * NP + m];
    v = 2.f * s - Tm2[i];
  }
  Tk[i] = v;
}

// ---------------------------------------------------------------------------
// Stage 1: fused Chebyshev conv + 10-step GL diffusion chain, one block per b.
// State lives entirely in LDS; L / polys stream from global (L2-resident).
// ---------------------------------------------------------------------------

// GL(in) -> out for this wave's two M-tiles.  t = (L@in - tanh(tanh(in@W0)@W1)) @ W2 + b
__device__ __forceinline__ void gl_apply(const float* __restrict__ Lp,
                                         const float* __restrict__ in,
                                         float* __restrict__ out,
                                         const float* __restrict__ w0,
                                         const float* __restrict__ w1,
                                         const float* __restrict__ w2,
                                         float* __restrict__ myscr,
                                         int wave, float tzb) {
  for (int t = wave; t < 20; t += 10) {
    const float* inT = in + t * 16 * ST;
    // spatial: o1 = L_tile @ in   (K = 320, 80 k-steps)
    v8f o1 = zero8();
#pragma unroll 4
    for (int kk = 0; kk < NP / 4; ++kk)
      o1 = wmma4(fragA(Lp + (t * 16) * NP + kk * 4, NP),
                 fragB(in + (kk * 4) * ST, ST), o1);
    // o3 = tanh(tanh(in @ w0) @ w1)
    v8f p = zero8();
#pragma unroll
    for (int kk = 0; kk < 4; ++kk)
      p = wmma4(fragA(inT + kk * 4, ST), fragB(w0 + kk * 4 * 16, 16), p);
    p = tanh8(p);
    stC(myscr, ST, p);
    v8f q = zero8();
#pragma unroll
    for (int kk = 0; kk < 4; ++kk)
      q = wmma4(fragA(myscr + kk * 4, ST), fragB(w1 + kk * 4 * 16, 16), q);
    q = tanh8(q);
    // t = (o1 - o3) @ w2 + bias
    v8f d = o1 - q;
    stC(myscr, ST, d);
    v8f r = zero8();
#pragma unroll
    for (int kk = 0; kk < 4; ++kk)
      r = wmma4(fragA(myscr + kk * 4, ST), fragB(w2 + kk * 4 * 16, 16), r);
#pragma unroll
    for (int v = 0; v < 8; ++v) r[v] += tzb;
    stC(out + t * 16 * ST, ST, r);
  }
}

__global__ __launch_bounds__(320) void gl_chain_kernel(
    const float* __restrict__ flow_x, const float* __restrict__ cou_w,
    const float* __restrict__ cheb_w, const float* __restrict__ cheb_b,
    const float* __restrict__ gl_out_w, const float* __restrict__ gl_fk_w,
    const float* __restrict__ gl_tz_w, const float* __restrict__ gl_tz_b,
    const float* __restrict__ Lp, const float* __restrict__ polysP,
    float* __restrict__ step0, float* __restrict__ step1) {
  __shared__ float sbuf[4][NP * ST];     // rotating state buffers
  __shared__ float xb[NP * 8];           // x_b padded [320 x 8]
  __shared__ float wgl[3][256];          // gl_out, gl_fk, gl_tz_w
  __shared__ float scr[10][16 * ST];     // per-wave 16x16 scratch

  const int tid  = threadIdx.x;
  const int wave = tid >> 5;
  const int b    = blockIdx.x;

  for (int i = tid; i < 4 * NP * ST; i += 320) (&sbuf[0][0])[i] = 0.f;
  for (int i = tid; i < NP * 8; i += 320) xb[i] = 0.f;
  if (tid < 256) {
    wgl[0][tid] = gl_out_w[tid];
    wgl[1][tid] = gl_fk_w[tid];
    wgl[2][tid] = gl_tz_w[tid];
  }
  __syncthreads();
  for (int i = tid; i < NN * CIN; i += 320) {
    int m = i / CIN, c = i - m * CIN;
    xb[m * 8 + c] = flow_x[(b * NN + m) * CIN + c];
  }
  __syncthreads();

  float* P = sbuf[0];  // prev2
  float* C = sbuf[1];  // cur
  float* T = sbuf[2];  // t
  float* S = sbuf[3];  // scratch

  // out0 = x @ cou_w -> P ;  C init = cheb_b
  {
    int m = tid;
    float x0 = xb[m*8+0], x1 = xb[m*8+1], x2 = xb[m*8+2], x3 = xb[m*8+3], x4 = xb[m*8+4];
#pragma unroll
    for (int h = 0; h < HH; ++h) {
      P[m*ST+h] = x0*cou_w[0*HH+h] + x1*cou_w[1*HH+h] + x2*cou_w[2*HH+h] +
                  x3*cou_w[3*HH+h] + x4*cou_w[4*HH+h];
      C[m*ST+h] = cheb_b[h];
    }
  }
  __syncthreads();

  // C += sum_k polys[k] @ (x @ cheb_w[k])
  for (int k = 0; k < KCH; ++k) {
    {
      int m = tid;
      float x0 = xb[m*8+0], x1 = xb[m*8+1], x2 = xb[m*8+2], x3 = xb[m*8+3], x4 = xb[m*8+4];
      const float* cw = cheb_w + k * CIN * HH;
#pragma unroll
      for (int h = 0; h < HH; ++h)
        S[m*ST+h] = x0*cw[h] + x1*cw[HH+h] + x2*cw[2*HH+h] + x3*cw[3*HH+h] + x4*cw[4*HH+h];
    }
    __syncthreads();
    const float* Pk = polysP + k * NP * NP;
    for (int t = wave; t < 20; t += 10) {
      v8f acc = ldC(C + t * 16 * ST, ST);
#pragma unroll 4
      for (int kk = 0; kk < NP / 4; ++kk)
        acc = wmma4(fragA(Pk + (t * 16) * NP + kk * 4, NP),
                    fragB(S + (kk * 4) * ST, ST), acc);
      stC(C + t * 16 * ST, ST, acc);
    }
    __syncthreads();
  }

  // RK-like diffusion: prev2,cur = cur, cur + 0.5 t + 0.5 GL(prev2 + 2t)
  const float tzb = gl_tz_b[tid & 15];
  for (int it = 0; it < 10; ++it) {
    gl_apply(Lp, C, T, wgl[0], wgl[1], wgl[2], scr[wave], wave, tzb);   // T = GL(C)
    __syncthreads();
    for (int i = tid; i < NP * HH; i += 320) {                          // S = P + 2T
      int m = i >> 4, h = i & 15;
      S[m*ST+h] = P[m*ST+h] + 2.f * T[m*ST+h];
    }
    __syncthreads();
    gl_apply(Lp, S, P, wgl[0], wgl[1], wgl[2], scr[wave], wave, tzb);   // P = GL(S) (P dead)
    __syncthreads();
    for (int i = tid; i < NP * HH; i += 320) {                          // S = C + .5T + .5P
      int m = i >> 4, h = i & 15;
      S[m*ST+h] = C[m*ST+h] + 0.5f * T[m*ST+h] + 0.5f * P[m*ST+h];
    }
    __syncthreads();
    float* oP = P; float* oT = T;
    P = C; C = S; T = oP; S = oT;   // rotate: prev2<=cur, cur<=new
  }

  for (int i = tid; i < NN * HH; i += 320) {
    int m = i >> 4, h = i & 15;
    step0[(b * NN + m) * HH + h] = P[m*ST+h];
    step1[(b * NN + m) * HH + h] = C[m*ST+h];
  }
}

// ---------------------------------------------------------------------------
// Stage 2: fused NL (LSTM-cell) chain, purely row-local. One wave per 16-row
// tile; whole 10-step chain in registers, guances stored straight to d_out.
// ---------------------------------------------------------------------------

__global__ __launch_bounds__(256) void nl_chain_kernel(
    const float* __restrict__ step0, const float* __restrict__ step1,
    const float* __restrict__ lstm_Wih, const float* __restrict__ lstm_b,
    const float* __restrict__ nl_w2, const float* __restrict__ c_w,
    const float* __restrict__ c_b, const float* __restrict__ c1_w,
    const float* __restrict__ c1_b, float* __restrict__ out) {
  __shared__ float sWih[16 * 128];
  __shared__ float sW2[32 * 16];
  __shared__ float sCw[256], sC1w[256];
  __shared__ float scrA[8][16 * ST];   // per-wave A-source scratch (16 x 16)
  __shared__ float scrH[8][16 * 33];   // per-wave hh scratch (16 x 32)

  const int tid = threadIdx.x, wave = tid >> 5, lane = tid & 31;
  for (int i = tid; i < 16 * 128; i += 256) sWih[i] = lstm_Wih[i];
  for (int i = tid; i < 32 * 16; i += 256) sW2[i] = nl_w2[i];
  sCw[tid]  = c_w[tid];
  sC1w[tid] = c1_w[tid];
  __syncthreads();

  const int tile = blockIdx.x * 8 + wave;   // 0..9823 exactly
  const int r0   = tile * 16;
  float* myA = scrA[wave];
  float* myH = scrH[wave];
  const int n = lane & 15;
  const float b_i0 = lstm_b[0*16+n], b_i1 = lstm_b[1*16+n];
  const float b_g0 = lstm_b[4*16+n], b_g1 = lstm_b[5*16+n];
  const float b_o0 = lstm_b[6*16+n], b_o1 = lstm_b[7*16+n];
  const float cb = c_b[n], c1b = c1_b[n];

  // step00 = step0 @ c_w + c_b ; step11 = step1 @ c1_w + c1_b
  v8f s00 = zero8(), s11 = zero8();
#pragma unroll
  for (int kk = 0; kk < 4; ++kk) {
    s00 = wmma4(fragA(step0 + r0 * 16 + kk * 4, 16), fragB(sCw  + kk * 64, 16), s00);
    s11 = wmma4(fragA(step1 + r0 * 16 + kk * 4, 16), fragB(sC1w + kk * 64, 16), s11);
  }
#pragma unroll
  for (int v = 0; v < 8; ++v) { s00[v] += cb; s11[v] += c1b; }
  v8f s1 = ldC(step1 + r0 * 16, 16);

  // NL(x): x pre-stored in myA.  gates = x@Wih + b (i,f,g,o order, f dead).
  auto nl_eval = [&]() -> v8f {
    v8f gi0 = zero8(), gi1 = zero8(), gg0 = zero8(), gg1 = zero8(),
        go0 = zero8(), go1 = zero8();
#pragma unroll
    for (int kk = 0; kk < 4; ++kk) {
      v2f a = fragA(myA + kk * 4, ST);
      gi0 = wmma4(a, fragB(sWih + kk * 4 * 128 +   0, 128), gi0);
      gi1 = wmma4(a, fragB(sWih + kk * 4 * 128 +  16, 128), gi1);
      gg0 = wmma4(a, fragB(sWih + kk * 4 * 128 +  64, 128), gg0);
      gg1 = wmma4(a, fragB(sWih + kk * 4 * 128 +  80, 128), gg1);
      go0 = wmma4(a, fragB(sWih + kk * 4 * 128 +  96, 128), go0);
      go1 = wmma4(a, fragB(sWih + kk * 4 * 128 + 112, 128), go1);
    }
    v8f h0, h1;
#pragma unroll
    for (int v = 0; v < 8; ++v) {
      float i0 = sigm(gi0[v] + b_i0), i1 = sigm(gi1[v] + b_i1);
      float g0 = tanhf(gg0[v] + b_g0), g1 = tanhf(gg1[v] + b_g1);
      float o0 = sigm(go0[v] + b_o0), o1 = sigm(go1[v] + b_o1);
      h0[v] = tanhf(o0 * tanhf(i0 * g0));
      h1[v] = tanhf(o1 * tanhf(i1 * g1));
    }
    stC(myH +  0, 33, h0);
    stC(myH + 16, 33, h1);
    v8f o = zero8();
#pragma unroll
    for (int kk = 0; kk < 8; ++kk)
      o = wmma4(fragA(myH + kk * 4, 33), fragB(sW2 + kk * 4 * 16, 16), o);
    return o;
  };

  auto store_g = [&](int q, v8f c) {
    const int mb = (lane >> 4) << 3;
#pragma unroll
    for (int v = 0; v < 8; ++v) {
      int r = r0 + mb + v;
      out[(r * 10 + q) * 16 + n] = c[v];
    }
  };

  v8f t, n2, g, prev2, u;
  stC(myA, ST, s11);
  t = nl_eval();
#pragma unroll
  for (int v = 0; v < 8; ++v) u[v] = s00[v] + 2.f * t[v];
  stC(myA, ST, u);
  n2 = nl_eval();
#pragma unroll
  for (int v = 0; v < 8; ++v) g[v] = 0.5f * t[v] + s1[v] + 0.5f * n2[v];
  store_g(0, g);
  prev2 = s11;

  for (int it = 1; it < 10; ++it) {
    stC(myA, ST, g);
    t = nl_eval();
#pragma unroll
    for (int v = 0; v < 8; ++v) u[v] = prev2[v] + 2.f * t[v];
    stC(myA, ST, u);
    n2 = nl_eval();
    v8f ng;
#pragma unroll
    for (int v = 0; v < 8; ++v) ng[v] = 0.5f * t[v] + g[v] + 0.5f * n2[v];
    prev2 = g; g = ng;
    store_g(it, g);
  }
}

// ---------------------------------------------------------------------------

extern "C" void kernel_launch(void* const* d_in, const int* in_sizes, int n_in,
                              void* d_out, int out_size, void* d_ws, size_t ws_size,
                              hipStream_t stream) {
  const float* flow_x   = (const float*)d_in[0];
  const float* node_emb = (const float*)d_in[1];
  const float* cheb_w   = (const float*)d_in[2];
  const float* cheb_b   = (const float*)d_in[3];
  const float* cou_w    = (const float*)d_in[4];
  const float* gl_out   = (const float*)d_in[5];
  const float* gl_fk    = (const float*)d_in[6];
  const float* gl_tz_w  = (const float*)d_in[7];
  const float* gl_tz_b  = (const float*)d_in[8];
  const float* lstm_Wih = (const float*)d_in[9];
  const float* lstm_b   = (const float*)d_in[10];
  const float* nl_w2    = (const float*)d_in[11];
  const float* c_w      = (const float*)d_in[12];
  const float* c_b      = (const float*)d_in[13];
  const float* c1_w     = (const float*)d_in[14];
  const float* c1_b     = (const float*)d_in[15];
  (void)in_sizes; (void)n_in; (void)out_size; (void)ws_size;

  float* ws    = (float*)d_ws;
  float* G     = ws;                       // 320*320
  float* dinv  = G + NP * NP;              // 320
  float* Lp    = dinv + NP;                // 320*320
  float* polys = Lp + NP * NP;             // 6 * 320*320
  float* step0 = polys + KCH * NP * NP;    // ROWS*16
  float* step1 = step0 + ROWS * HH;        // ROWS*16   (total ~23.4 MB)

  const int g2d = (NP * NP + 255) / 256;   // 400 blocks
  graph_kernel<<<g2d, 256, 0, stream>>>(node_emb, G);
  degree_kernel<<<1, NP, 0, stream>>>(G, dinv);
  lap_kernel<<<g2d, 256, 0, stream>>>(G, dinv, Lp, polys, polys + NP * NP);
  for (int k = 2; k < KCH; ++k)
    cheb_kernel<<<g2d, 256, 0, stream>>>(Lp, polys + (k - 1) * NP * NP,
                                         polys + (k - 2) * NP * NP,
                                         polys + k * NP * NP);

  gl_chain_kernel<<<BB, 320, 0, stream>>>(flow_x, cou_w, cheb_w, cheb_b,
                                          gl_out, gl_fk, gl_tz_w, gl_tz_b,
                                          Lp, polys, step0, step1);

  nl_chain_kernel<<<(ROWS / 16) / 8, 256, 0, stream>>>(
      step0, step1, lstm_Wih, lstm_b, nl_w2, c_w, c_b, c1_w, c1_b,
      (float*)d_out);
}